// TrainableGeneSetLayer_54443005444402
// MI455X (gfx1250) — compile-verified
//
#include <hip/hip_runtime.h>
#include <hip/hip_bf16.h>

#define G_GENES 8192
#define N_SETS  256
#define N_BATCH 32

typedef __attribute__((ext_vector_type(2))) float v2f;
typedef __attribute__((ext_vector_type(8))) float v8f;

// p = ind^0.25 ; neg test (ind < 0.1)  <=>  (p < 0.1^0.25)
#define PNEG_THRESH 0.56234132519f

// ---------------------------------------------------------------------------
// Kernel 1: indicators. One block per set m.
//   sig = sigmoid(W[m,:]) staged in LDS; avg = mean; ind = where(sig<avg*0.3, sig*0.01, sig)
//   pT[g*256+m] = ind^0.25   (transposed for WMMA A-operand coalescing)
//   sneg[m]     = count(ind < 0.1)   (permutation-independent)
// ---------------------------------------------------------------------------
__global__ __launch_bounds__(256) void k_indicators(const float* __restrict__ W,
                                                    float* __restrict__ pT,
                                                    float* __restrict__ sneg) {
    __shared__ float sig[G_GENES];
    __shared__ float red[256];
    const int m = blockIdx.x;
    const int t = threadIdx.x;

    float acc = 0.f;
    for (int i = 0; i < G_GENES / 256; ++i) {
        int g = t + i * 256;
        float x = W[(size_t)m * G_GENES + g];
        float s = 1.0f / (1.0f + __expf(-x));
        sig[g] = s;
        acc += s;
    }
    red[t] = acc;
    __syncthreads();
    for (int off = 128; off > 0; off >>= 1) {
        if (t < off) red[t] += red[t + off];
        __syncthreads();
    }
    const float thr = red[0] * (1.0f / G_GENES) * 0.3f;
    __syncthreads();

    float negc = 0.f;
    for (int i = 0; i < G_GENES / 256; ++i) {
        int g = t + i * 256;
        float s = sig[g];
        float indv = (s < thr) ? s * 0.01f : s;
        pT[(size_t)g * N_SETS + m] = sqrtf(sqrtf(indv));   // ind^0.25
        negc += (indv < 0.1f) ? 1.0f : 0.0f;
    }
    red[t] = negc;
    __syncthreads();
    for (int off = 128; off > 0; off >>= 1) {
        if (t < off) red[t] += red[t + off];
        __syncthreads();
    }
    if (t == 0) sneg[m] = red[0];
}

// ---------------------------------------------------------------------------
// Kernel 2: scatter permutation statistics into (G x B) matrices (WMMA B-operand
// layout: lane index runs along b, contiguous).
//   g = S[b,j]; a = R[b,g]^0.25; wj = (G - j)
//   Ut[g*32+b] = wj*a ; Vt[g*32+b] = a ; Wt[g*32+b] = wj
// ---------------------------------------------------------------------------
__global__ __launch_bounds__(256) void k_scatter(const float* __restrict__ R,
                                                 const int*   __restrict__ S,
                                                 float* __restrict__ Ut,
                                                 float* __restrict__ Vt,
                                                 float* __restrict__ Wt) {
    int idx = blockIdx.x * 256 + threadIdx.x;     // 0 .. B*G-1
    int b = idx >> 13;                            // /8192
    int j = idx & (G_GENES - 1);
    int g = S[idx];
    float r = R[(size_t)b * G_GENES + g];
    float a = sqrtf(sqrtf(r));                    // R_sorted^0.25
    float wj = (float)(G_GENES - j);              // inclusive-cumsum weight
    size_t o = (size_t)g * N_BATCH + b;
    Ut[o] = wj * a;
    Vt[o] = a;
    Wt[o] = wj;
}

// ---------------------------------------------------------------------------
// Kernel 3: three fused f32 WMMA GEMMs + epilogue.
//   D tile = 16 m-rows x 16 b-cols. Grid: 16 m-tiles x 2 b-tiles = 32 blocks.
//   8 waves/block split K=8192; per 4-K step: 3x v_wmma_f32_16x16x4_f32
//   (Apos: p x U, spos: p x V, Aneg: neg(p) x W). Cross-wave reduce in LDS.
// ---------------------------------------------------------------------------
__global__ __launch_bounds__(256) void k_gemm(const float* __restrict__ pT,
                                              const float* __restrict__ Ut,
                                              const float* __restrict__ Vt,
                                              const float* __restrict__ Wt,
                                              const float* __restrict__ sneg,
                                              float* __restrict__ out) {
    __shared__ float red[3 * 8 * 256];            // 24 KB: 3 products x 8 waves x 16x16

    const int mtile = blockIdx.x & 15;
    const int btile = blockIdx.x >> 4;
    const int m0 = mtile * 16;
    const int b0 = btile * 16;

    const int wave = threadIdx.x >> 5;            // 0..7
    const int lane = threadIdx.x & 31;
    const int l15  = lane & 15;
    const int rowm = m0 + l15;                    // A operand: row = m (lanes 0-15 / 16-31)
    const int colb = b0 + l15;                    // B operand: col = b
    const int kh   = (lane >> 4) << 1;            // K sub-offset: 0 (lo half) / 2 (hi half)

    v8f cApos = {};
    v8f cSpos = {};
    v8f cAneg = {};

    const int kbeg = wave * (G_GENES / 8);
    const int kend = kbeg + (G_GENES / 8);
#pragma unroll 4
    for (int k = kbeg; k < kend; k += 4) {
        const int ka = k + kh;
        v2f aP, bU, bV, bW, aN;
        aP.x = pT[(size_t)ka       * N_SETS + rowm];
        aP.y = pT[(size_t)(ka + 1) * N_SETS + rowm];
        bU.x = Ut[(size_t)ka       * N_BATCH + colb];
        bU.y = Ut[(size_t)(ka + 1) * N_BATCH + colb];
        bV.x = Vt[(size_t)ka       * N_BATCH + colb];
        bV.y = Vt[(size_t)(ka + 1) * N_BATCH + colb];
        bW.x = Wt[(size_t)ka       * N_BATCH + colb];
        bW.y = Wt[(size_t)(ka + 1) * N_BATCH + colb];
        aN.x = (aP.x < PNEG_THRESH) ? 1.0f : 0.0f;
        aN.y = (aP.y < PNEG_THRESH) ? 1.0f : 0.0f;

        cApos = __builtin_amdgcn_wmma_f32_16x16x4_f32(false, aP, false, bU,
                                                      (short)0, cApos, false, false);
        cSpos = __builtin_amdgcn_wmma_f32_16x16x4_f32(false, aP, false, bV,
                                                      (short)0, cSpos, false, false);
        cAneg = __builtin_amdgcn_wmma_f32_16x16x4_f32(false, aN, false, bW,
                                                      (short)0, cAneg, false, false);
    }

    // C layout: component i, lanes 0-15 -> row i, col lane; lanes 16-31 -> row i+8
    {
        const int hi = lane >> 4;
        const int c  = lane & 15;
#pragma unroll
        for (int i = 0; i < 8; ++i) {
            int r = i + 8 * hi;
            red[0 * 2048 + wave * 256 + r * 16 + c] = cApos[i];
            red[1 * 2048 + wave * 256 + r * 16 + c] = cSpos[i];
            red[2 * 2048 + wave * 256 + r * 16 + c] = cAneg[i];
        }
    }
    __syncthreads();

    // Epilogue: one thread per output element of the 16x16 tile.
    {
        const int e = threadIdx.x;                // 0..255 = r*16 + c
        const int r = e >> 4;
        const int c = e & 15;
        float Apos = 0.f, Spos = 0.f, Aneg = 0.f;
#pragma unroll
        for (int w = 0; w < 8; ++w) {
            Apos += red[0 * 2048 + w * 256 + e];
            Spos += red[1 * 2048 + w * 256 + e];
            Aneg += red[2 * 2048 + w * 256 + e];
        }
        const int m = m0 + r;
        const int b = b0 + c;
        const float sn = sneg[m];
        const float negterm = (sn > 1e-8f) ? (Aneg / (sn + 1e-10f)) : 0.0f;
        out[(size_t)b * N_SETS + m] =
            (Apos / (Spos + 1e-10f) - negterm) * (1.0f / (float)G_GENES);
    }
}

// ---------------------------------------------------------------------------
extern "C" void kernel_launch(void* const* d_in, const int* in_sizes, int n_in,
                              void* d_out, int out_size, void* d_ws, size_t ws_size,
                              hipStream_t stream) {
    const float* R = (const float*)d_in[0];       // (32, 8192) f32
    const int*   S = (const int*)  d_in[1];       // (32, 8192) i32
    const float* W = (const float*)d_in[2];       // (256, 8192) f32
    float* out = (float*)d_out;                   // (32, 256) f32

    char* ws = (char*)d_ws;
    const size_t szPT = (size_t)G_GENES * N_SETS  * sizeof(float);  // 8 MB
    const size_t szGB = (size_t)G_GENES * N_BATCH * sizeof(float);  // 1 MB
    float* pT   = (float*)(ws);
    float* Ut   = (float*)(ws + szPT);
    float* Vt   = (float*)(ws + szPT + szGB);
    float* Wt   = (float*)(ws + szPT + 2 * szGB);
    float* sneg = (float*)(ws + szPT + 3 * szGB);

    k_indicators<<<N_SETS, 256, 0, stream>>>(W, pT, sneg);
    k_scatter<<<(N_BATCH * G_GENES) / 256, 256, 0, stream>>>(R, S, Ut, Vt, Wt);
    k_gemm<<<32, 256, 0, stream>>>(pT, Ut, Vt, Wt, sneg, out);
}